// Prgrph_ending_classifier_27066883899597
// MI455X (gfx1250) — compile-verified
//
#include <hip/hip_runtime.h>
#include <hip/hip_bf16.h>
#include <math.h>

// Shapes: B=2048, S=32, D_ENC=2048, N_ENT=64, D_ENT=1024
#define B_SZ   2048
#define S_SZ   32
#define DENC   2048
#define NENT   64
#define DENT   1024

typedef __attribute__((ext_vector_type(2))) float v2f;
typedef __attribute__((ext_vector_type(4))) float v4f;
typedef __attribute__((ext_vector_type(8))) float v8f;

__device__ __forceinline__ float wave_sum32(float v) {
    v += __shfl_xor(v, 16, 32);
    v += __shfl_xor(v, 8, 32);
    v += __shfl_xor(v, 4, 32);
    v += __shfl_xor(v, 2, 32);
    v += __shfl_xor(v, 1, 32);
    return v;
}
__device__ __forceinline__ float wave_max32(float v) {
    v = fmaxf(v, __shfl_xor(v, 16, 32));
    v = fmaxf(v, __shfl_xor(v, 8, 32));
    v = fmaxf(v, __shfl_xor(v, 4, 32));
    v = fmaxf(v, __shfl_xor(v, 2, 32));
    v = fmaxf(v, __shfl_xor(v, 1, 32));
    return v;
}
// reduce across lanes 0..15 / 16..31 independently (xor masks < 16)
__device__ __forceinline__ float half_sum16(float v) {
    v += __shfl_xor(v, 8, 32);
    v += __shfl_xor(v, 4, 32);
    v += __shfl_xor(v, 2, 32);
    v += __shfl_xor(v, 1, 32);
    return v;
}

// ---------------------------------------------------------------- K0: zero ms
__global__ void zero_ms_kernel(float* __restrict__ ms) {
    ms[blockIdx.x * 256 + threadIdx.x] = 0.0f;   // <<<4,256>>> zeroes 1024 floats
}

// ------------------------------------------------- K1: masked_sum[e] (1024)
// ms[e] = sum_{b,n} entities[b,n,e] * mask[b,n]; 64 blocks x 32 batches each.
// Each thread owns 4 contiguous e -> one global_load_b128 per (b,n) row.
__global__ __launch_bounds__(256) void masked_sum_kernel(
        const float* __restrict__ ent, const unsigned char* __restrict__ mask,
        float* __restrict__ ms) {
    const int t  = threadIdx.x;
    const int e0 = t * 4;
    const int b0 = blockIdx.x * 32;
    v4f acc = {0.f, 0.f, 0.f, 0.f};
    for (int bl = 0; bl < 32; ++bl) {
        const int b = b0 + bl;
        const size_t ebase = (size_t)b * NENT * DENT;
        for (int n = 0; n < NENT; ++n) {
            if (mask[b * NENT + n]) {
                acc += *(const v4f*)(ent + ebase + (size_t)n * DENT + e0);
            }
        }
    }
#pragma unroll
    for (int j = 0; j < 4; ++j) atomicAdd(&ms[e0 + j], acc[j]);
}

// ------------------------------ K2: posemb + self-attn -> attn_hiddens row
// One block per batch. x tile (32x2048 f32 = 256KB) staged in LDS (<=320KB).
// Each thread owns 8 contiguous d -> b128 global loads/stores.
__global__ __launch_bounds__(256) void attn_kernel(
        const float* __restrict__ enc, float* __restrict__ Aws) {
    __shared__ float xs[S_SZ * DENC];   // 256 KB
    __shared__ float lg[S_SZ];
    __shared__ float cf[S_SZ];

    const int b    = blockIdx.x;
    const int t    = threadIdx.x;
    const int lane = t & 31;
    const int d0   = t * 8;
    const size_t base = (size_t)b * S_SZ * DENC;

    if (t < S_SZ) lg[t] = 0.0f;
    __syncthreads();

    // per-column sinusoid frequency: 200^(-d/D) = exp(-d*ln(200)/D)
    const float c0 = -5.2983173665480363f / (float)DENC;  // -ln(200)/D
    float freq[8], q[8];
#pragma unroll
    for (int j = 0; j < 8; ++j) freq[j] = __expf(c0 * (float)(d0 + j));

    {   // row 31 (query): load, add posemb, stash in LDS + regs
        v4f e0 = *(const v4f*)(enc + base + 31 * DENC + d0);
        v4f e1 = *(const v4f*)(enc + base + 31 * DENC + d0 + 4);
#pragma unroll
        for (int j = 0; j < 4; ++j) {
            q[j]     = e0[j] + __sinf(31.0f * freq[j]);
            q[4 + j] = e1[j] + __sinf(31.0f * freq[4 + j]);
        }
        v4f q0 = {q[0], q[1], q[2], q[3]};
        v4f q1 = {q[4], q[5], q[6], q[7]};
        *(v4f*)(&xs[31 * DENC + d0])     = q0;
        *(v4f*)(&xs[31 * DENC + d0 + 4]) = q1;
        float a = 0.f;
#pragma unroll
        for (int j = 0; j < 8; ++j) a += q[j] * q[j];
        a = wave_sum32(a);
        if (lane == 0) atomicAdd(&lg[31], a);   // logits[31] = |q|^2
    }
    for (int s = 0; s < 31; ++s) {
        const float sf = (float)s;
        v4f e0 = *(const v4f*)(enc + base + (size_t)s * DENC + d0);
        v4f e1 = *(const v4f*)(enc + base + (size_t)s * DENC + d0 + 4);
        float vv[8];
#pragma unroll
        for (int j = 0; j < 4; ++j) {
            vv[j]     = e0[j] + __sinf(sf * freq[j]);
            vv[4 + j] = e1[j] + __sinf(sf * freq[4 + j]);
        }
        v4f x0 = {vv[0], vv[1], vv[2], vv[3]};
        v4f x1 = {vv[4], vv[5], vv[6], vv[7]};
        *(v4f*)(&xs[s * DENC + d0])     = x0;
        *(v4f*)(&xs[s * DENC + d0 + 4]) = x1;
        float a = 0.f;
#pragma unroll
        for (int j = 0; j < 8; ++j) a += vv[j] * q[j];
        a = wave_sum32(a);
        if (lane == 0) atomicAdd(&lg[s], a);
    }
    __syncthreads();

    if (t < 32) {   // softmax over 32 logits, one wave
        const float v = lg[t];
        const float m = wave_max32(v);
        const float e = __expf(v - m);
        const float s = wave_sum32(e);
        cf[t] = e / s;
    }
    __syncthreads();

    // attn_hiddens[b,d] = sum_s cf[s] * x[s,d]
    float cfr[S_SZ];
#pragma unroll
    for (int s = 0; s < S_SZ; ++s) cfr[s] = cf[s];
    v4f acc0 = {0.f, 0.f, 0.f, 0.f};
    v4f acc1 = {0.f, 0.f, 0.f, 0.f};
#pragma unroll
    for (int s = 0; s < S_SZ; ++s) {
        v4f x0 = *(const v4f*)(&xs[s * DENC + d0]);
        v4f x1 = *(const v4f*)(&xs[s * DENC + d0 + 4]);
        acc0 += x0 * cfr[s];
        acc1 += x1 * cfr[s];
    }
    *(v4f*)(Aws + (size_t)b * DENC + d0)     = acc0;
    *(v4f*)(Aws + (size_t)b * DENC + d0 + 4) = acc1;
}

// ---------------- K3: qW = A @ W via V_WMMA_F32_16X16X4_F32, fused epilogue
// out[b] = sigmoid( sum_n qW[b,n] * ms[n] * dw[n] + db )
// 128 blocks x 16-row A strips; A strip in LDS with pad (stride 2052) so the
// per-fragment ds_load_b64 (lanes 0-15 same K, stride-2052 rows) is conflict
// free. 8 waves x 8 N-tiles of 16 each cover N=1024.
#define APAD 2052
__global__ __launch_bounds__(256) void gemm_fuse_kernel(
        const float* __restrict__ Aws, const float* __restrict__ W,
        const float* __restrict__ ms, const float* __restrict__ dw,
        const float* __restrict__ db, float* __restrict__ out) {
    __shared__ float As[16 * APAD];     // ~128 KB
    __shared__ float outred[16];

    const int t  = threadIdx.x;
    const int m0 = blockIdx.x * 16;

    // cooperative b128 A-strip load (columns 4-aligned so LDS stores are b128)
    for (int i = t; i < 16 * DENC / 4; i += 256) {
        const int el = i * 4;
        const int r = el >> 11, c = el & (DENC - 1);
        *(v4f*)(&As[r * APAD + c]) = *(const v4f*)(Aws + (size_t)(m0 + r) * DENC + c);
    }
    if (t < 16) outred[t] = 0.0f;
    __syncthreads();

    const int w    = t >> 5;
    const int lane = t & 31;
    const int half = lane >> 4;     // 0: K=k0..k0+1 ; 1: K=k0+2..k0+3 (A layout)
    const int l16  = lane & 15;

    float outacc[8] = {0.f, 0.f, 0.f, 0.f, 0.f, 0.f, 0.f, 0.f};

    for (int ntb = 0; ntb < 8; ++ntb) {
        const int nt = ntb * 8 + w;          // N-tile 0..63
        const int n  = nt * 16 + l16;        // this lane's column
        v8f c = {0.f, 0.f, 0.f, 0.f, 0.f, 0.f, 0.f, 0.f};
        for (int k0 = 0; k0 < DENC; k0 += 4) {
            const int ka = k0 + half * 2;
            v2f a, bfrag;
            a.x = As[l16 * APAD + ka];       // A[M=l16, K=ka]
            a.y = As[l16 * APAD + ka + 1];   // A[M=l16, K=ka+1]
            bfrag.x = W[(size_t)ka * DENT + n];        // B[K=ka,   N=n]
            bfrag.y = W[(size_t)(ka + 1) * DENT + n];  // B[K=ka+1, N=n]
            c = __builtin_amdgcn_wmma_f32_16x16x4_f32(
                    false, a, false, bfrag, (short)0, c, false, false);
        }
        const float f = ms[n] * dw[n];       // fold masked_sum * dense_w
#pragma unroll
        for (int r = 0; r < 8; ++r) outacc[r] += c[r] * f;
    }

    // C layout: VGPR r, lanes0-15 -> M=m0+r ; lanes16-31 -> M=m0+8+r.
#pragma unroll
    for (int r = 0; r < 8; ++r) {
        const float v = half_sum16(outacc[r]);
        if (l16 == 0) atomicAdd(&outred[r + half * 8], v);
    }
    __syncthreads();
    if (t < 16) {
        const float x = outred[t] + db[0];
        out[m0 + t] = 1.0f / (1.0f + __expf(-x));
    }
}

// ---------------------------------------------------------------- launcher
extern "C" void kernel_launch(void* const* d_in, const int* in_sizes, int n_in,
                              void* d_out, int out_size, void* d_ws, size_t ws_size,
                              hipStream_t stream) {
    (void)in_sizes; (void)n_in; (void)out_size; (void)ws_size;
    const float*         enc  = (const float*)d_in[0];          // (2048,32,2048)
    const float*         ent  = (const float*)d_in[1];          // (2048,64,1024)
    const unsigned char* mask = (const unsigned char*)d_in[2];  // (2048,64) bool
    const float*         W    = (const float*)d_in[3];          // (2048,1024)
    const float*         dw   = (const float*)d_in[4];          // (1024,1)
    const float*         db   = (const float*)d_in[5];          // (1,)
    float* out = (float*)d_out;                                 // (2048,1)

    float* wsf = (float*)d_ws;
    float* ms  = wsf;            // 1024 floats: masked entity sum
    float* Aws = wsf + 1024;     // 2048*2048 floats: attn_hiddens (16 MB)

    zero_ms_kernel   <<<4,    256, 0, stream>>>(ms);
    masked_sum_kernel<<<64,   256, 0, stream>>>(ent, mask, ms);
    attn_kernel      <<<2048, 256, 0, stream>>>(enc, Aws);
    gemm_fuse_kernel <<<128,  256, 0, stream>>>(Aws, W, ms, dw, db, out);
}